// PrototypeRetriever_35974646071931
// MI455X (gfx1250) — compile-verified
//
#include <hip/hip_runtime.h>

#define B_Q 32
#define D_DIM 512
#define TOPK 16
#define N_BANK 100000
#define TILE_N 32
#define TILES_PER_CHUNK 8
#define CHUNK (TILE_N * TILES_PER_CHUNK)          /* 256 */
#define N_CHUNKS ((N_BANK + CHUNK - 1) / CHUNK)   /* 391 */

typedef float v2f __attribute__((ext_vector_type(2)));
typedef float v8f __attribute__((ext_vector_type(8)));
typedef unsigned int v4u __attribute__((ext_vector_type(4)));
typedef int v4i __attribute__((ext_vector_type(4)));
typedef int v8i __attribute__((ext_vector_type(8)));

#if defined(__AMDGCN__) && defined(__has_builtin)
#if __has_builtin(__builtin_amdgcn_tensor_load_to_lds) && __has_builtin(__builtin_amdgcn_s_wait_tensorcnt)
#define USE_TDM 1
#endif
#endif

// Padded LDS layout: row stride 516 dwords; 2-dword pad after every 256 dwords
// (matches TDM pad_interval=256dw / pad_amount=2dw so DMA'd tiles land in this
// exact layout). 516 % 64 == 4 -> 16 lanes reading 16 consecutive rows at the
// same k hit 16 distinct banks (conflict-free WMMA fragment loads).
__device__ __forceinline__ int lpad(int row, int k) {
    return row * 516 + k + ((k >> 8) << 1);
}

__device__ __forceinline__ v8f wmma_f32_k4(v2f a, v2f b, v8f c) {
#if defined(__AMDGCN__)
    return __builtin_amdgcn_wmma_f32_16x16x4_f32(false, a, false, b, (short)0, c, false, false);
#else
    return c;
#endif
}

// Deterministic top-16 insert: fully unrolled argmin-replace (registers only).
__device__ __forceinline__ void insert16(float (&tv)[16], int (&ti)[16], float v, int id) {
    float vmin = tv[0];
    int pmin = 0;
#pragma unroll
    for (int s = 1; s < 16; s++) {
        if (tv[s] < vmin) { vmin = tv[s]; pmin = s; }
    }
    if (v > vmin) {
#pragma unroll
        for (int s = 0; s < 16; s++) {
            if (s == pmin) { tv[s] = v; ti[s] = id; }
        }
    }
}

// ---------------- Kernel 1: global average pool + l2norm -> qg [32,512] ----
__global__ void k_gap(const float* __restrict__ q, float* __restrict__ out_qg) {
    __shared__ float sv[512];
    __shared__ float red[256];
    int b = blockIdx.x, t = threadIdx.x;
    const float4* p0 = reinterpret_cast<const float4*>(q + ((size_t)b * 512 + t) * 196);
    const float4* p1 = reinterpret_cast<const float4*>(q + ((size_t)b * 512 + t + 256) * 196);
    float m0 = 0.f, m1 = 0.f;
    for (int i = 0; i < 49; i++) {
        float4 a = p0[i], c = p1[i];
        m0 += (a.x + a.y) + (a.z + a.w);
        m1 += (c.x + c.y) + (c.z + c.w);
    }
    m0 *= (1.0f / 196.0f);
    m1 *= (1.0f / 196.0f);
    sv[t] = m0;
    sv[t + 256] = m1;
    red[t] = m0 * m0 + m1 * m1;
    __syncthreads();
    for (int s = 128; s > 0; s >>= 1) {
        if (t < s) red[t] += red[t + s];
        __syncthreads();
    }
    float inv = 1.0f / fmaxf(sqrtf(red[0]), 1e-12f);
    out_qg[b * 512 + t] = sv[t] * inv;
    out_qg[b * 512 + t + 256] = sv[t + 256] * inv;
}

// ---------------- Kernel 2: MLP (GEMM->GELU->GEMM) + l2norm -> pq ----------
__global__ void __launch_bounds__(256) k_mlp(const float* __restrict__ qg,
                                             const float* __restrict__ W1,
                                             const float* __restrict__ b1,
                                             const float* __restrict__ W2,
                                             const float* __restrict__ b2,
                                             float* __restrict__ out_pq) {
    __shared__ float A[32 * 516];  // qg, later raw pq
    __shared__ float H[32 * 516];  // hidden
    __shared__ float red[256];
    int t = threadIdx.x;
    int lane = t & 31, w = t >> 5;

    for (int i = t; i < 32 * 512; i += 256) A[lpad(i >> 9, i & 511)] = qg[i];
    __syncthreads();

    // GEMM1: h = gelu(qg @ W1^T + b1). 64 output tiles (2M x 32N) over 8 waves.
    for (int tile = w; tile < 64; tile += 8) {
        int m = tile & 1, n0 = (tile >> 1) << 4;
        v8f c = {};
        int arow = (m << 4) + (lane & 15);
        int koff = (lane >> 4) << 1;
        const float* wrow = W1 + (size_t)(n0 + (lane & 15)) * 512 + koff;
        for (int k = 0; k < 512; k += 4) {
            v2f a = *(const v2f*)&A[lpad(arow, k + koff)];
            v2f b = *(const v2f*)(wrow + k);
            c = wmma_f32_k4(a, b, c);
        }
        int col = n0 + (lane & 15);
        float bias = b1[col];
        int rbase = (m << 4) + ((lane >> 4) << 3);
#pragma unroll
        for (int r = 0; r < 8; r++) {
            float x = c[r] + bias;
            float g = 0.5f * x * (1.0f + erff(x * 0.70710678118654752440f));
            H[lpad(rbase + r, col)] = g;
        }
    }
    __syncthreads();

    // GEMM2: pq_raw = h @ W2^T + b2 (written into A)
    for (int tile = w; tile < 64; tile += 8) {
        int m = tile & 1, n0 = (tile >> 1) << 4;
        v8f c = {};
        int arow = (m << 4) + (lane & 15);
        int koff = (lane >> 4) << 1;
        const float* wrow = W2 + (size_t)(n0 + (lane & 15)) * 512 + koff;
        for (int k = 0; k < 512; k += 4) {
            v2f a = *(const v2f*)&H[lpad(arow, k + koff)];
            v2f b = *(const v2f*)(wrow + k);
            c = wmma_f32_k4(a, b, c);
        }
        int col = n0 + (lane & 15);
        float bias = b2[col];
        int rbase = (m << 4) + ((lane >> 4) << 3);
#pragma unroll
        for (int r = 0; r < 8; r++) A[lpad(rbase + r, col)] = c[r] + bias;
    }
    __syncthreads();

    // l2norm per row
    for (int r = 0; r < 32; r++) {
        float v0 = A[lpad(r, t)], v1 = A[lpad(r, t + 256)];
        red[t] = v0 * v0 + v1 * v1;
        __syncthreads();
        for (int s = 128; s > 0; s >>= 1) {
            if (t < s) red[t] += red[t + s];
            __syncthreads();
        }
        float inv = 1.0f / fmaxf(sqrtf(red[0]), 1e-12f);
        out_pq[r * 512 + t] = v0 * inv;
        out_pq[r * 512 + t + 256] = v1 * inv;
        __syncthreads();
    }
}

// ---------------- Kernel 3: bank similarities + per-chunk top-16 -----------
// 128 threads (4 waves), ~145 KB LDS -> 2 blocks/WGP so one block's TDM DMA
// overlaps the other block's WMMA compute. 391 blocks saturate HBM streams.
__global__ void __launch_bounds__(128) k_sim(const float* __restrict__ pq,
                                             const float* __restrict__ bank,
                                             float* __restrict__ pvals,
                                             int* __restrict__ pidx) {
    __shared__ float PQ[32 * 516];
    __shared__ float BT[TILE_N * 516];
    __shared__ float NRM[TILE_N];
    __shared__ float PART[TILE_N][4];
    __shared__ float SIMS[4][16][16];
    __shared__ float TOPV[4][16][16];
    __shared__ int TOPI[4][16][16];

    int t = threadIdx.x, lane = t & 31, w = t >> 5;
    int m = w & 1, nsub = w >> 1;  // m in {0,1}, nsub in {0,1}

    for (int i = t; i < 32 * 512; i += 128) PQ[lpad(i >> 9, i & 511)] = pq[i];

    float tv[16];
    int ti[16];
#pragma unroll
    for (int s = 0; s < 16; s++) { tv[s] = -__builtin_inff(); ti[s] = 0; }

    int base = blockIdx.x * CHUNK;
    for (int it = 0; it < TILES_PER_CHUNK; it++) {
        int tb = base + it * TILE_N;
        if (tb >= N_BANK) break;  // uniform across block
        __syncthreads();          // prior tile fully consumed

#ifdef USE_TDM
        // Tensor Data Mover: DMA 32x512 f32 tile (row padded 2 dw / 256 dw) to LDS.
        if (t < 32) {
            unsigned long long ga =
                (unsigned long long)(uintptr_t)(bank + (size_t)tb * 512);
            unsigned int lds = (unsigned int)(uintptr_t)(&BT[0]);
            int rows = N_BANK - tb;
            if (rows > TILE_N) rows = TILE_N;
            v4u g0;
            g0[0] = 1u;                         // count=1 valid descriptor
            g0[1] = lds;                        // lds_addr
            g0[2] = (unsigned int)ga;           // global_addr[31:0]
            g0[3] = ((unsigned int)(ga >> 32) & 0x1FFFFFFu) | (2u << 30);  // addr hi | type=2
            v8i g1;
            g1[0] = (int)((2u << 16)     // data_size = 4B
                          | (1u << 20)   // pad_enable
                          | (7u << 22)   // pad_interval = 256 dwords
                          | (1u << 25)); // pad_amount = 2 dwords
            g1[1] = (int)(512u << 16);                        // tensor_dim0 = 512
            g1[2] = (int)(((unsigned)rows & 0xFFFFu) << 16);  // tensor_dim1 lo
            g1[3] = (int)((512u << 16) | (((unsigned)rows >> 16) & 0xFFFFu));  // tile_dim0 | dim1 hi
            g1[4] = TILE_N;  // tile_dim1 (OOB rows read as zero)
            g1[5] = 512;     // tensor_dim0_stride
            g1[6] = 0;
            g1[7] = 0;
            v4i gz = {0, 0, 0, 0};
#if __clang_major__ >= 23
            v8i gz8 = {0, 0, 0, 0, 0, 0, 0, 0};
            __builtin_amdgcn_tensor_load_to_lds(g0, g1, gz, gz, gz8, 0);
#else
            __builtin_amdgcn_tensor_load_to_lds(g0, g1, gz, gz, 0);
#endif
            __builtin_amdgcn_s_wait_tensorcnt(0);
        }
#else
        for (int i = t; i < TILE_N * 256; i += 128) {
            int r = i >> 8, k = (i & 255) << 1;
            int gr = tb + r;
            if (gr < N_BANK) {
                v2f v = *(const v2f*)(bank + (size_t)gr * 512 + k);
                *(v2f*)&BT[lpad(r, k)] = v;
            } else {
                v2f z = {0.f, 0.f};
                *(v2f*)&BT[lpad(r, k)] = z;
            }
        }
#endif
        __syncthreads();

        // bank-row norms: 4 deterministic partials per row, combined in order
        {
            int r = t >> 2, qk = (t & 3) << 7;
            float s = 0.f;
            for (int k = 0; k < 128; k++) {
                float x = BT[lpad(r, qk + k)];
                s += x * x;
            }
            PART[r][t & 3] = s;
        }
        __syncthreads();
        if (t < TILE_N) NRM[t] = (PART[t][0] + PART[t][1]) + (PART[t][2] + PART[t][3]);
        __syncthreads();

        // 16x16 f32 WMMA over K=512 (128 x v_wmma_f32_16x16x4_f32 per wave)
        v8f c = {};
        int arow = (m << 4) + (lane & 15);
        int brow = (nsub << 4) + (lane & 15);
        int koff = (lane >> 4) << 1;
        for (int k = 0; k < 512; k += 4) {
            v2f a = *(const v2f*)&PQ[lpad(arow, k + koff)];
            v2f b = *(const v2f*)&BT[lpad(brow, k + koff)];
            c = wmma_f32_k4(a, b, c);
        }
        int n_local = lane & 15;
        int grow = tb + (nsub << 4) + n_local;
        float inv = rsqrtf(fmaxf(NRM[(nsub << 4) + n_local], 1e-24f));
        bool valid = grow < N_BANK;
        int qb = (lane >> 4) << 3;
#pragma unroll
        for (int r = 0; r < 8; r++)
            SIMS[w][qb + r][n_local] = valid ? c[r] * inv : -__builtin_inff();
        __syncthreads();

        // per-lane top-16 (lane q of each wave owns query (m*16+q))
        if (lane < 16) {
#pragma unroll
            for (int j = 0; j < 16; j++) {
                float val = SIMS[w][lane][j];
                int idx = tb + (nsub << 4) + j;
                insert16(tv, ti, val, idx);
            }
        }
    }
    __syncthreads();

    if (lane < 16) {
#pragma unroll
        for (int s = 0; s < 16; s++) {
            TOPV[w][lane][s] = tv[s];
            TOPI[w][lane][s] = ti[s];
        }
    }
    __syncthreads();

    // merge the 2 wave-lists per query, write per-chunk partial top-16
    if (t < 32) {
        int q = t, qm = q >> 4, ql = q & 15;
        float mv[16];
        int mi[16];
#pragma unroll
        for (int s = 0; s < 16; s++) {
            mv[s] = TOPV[qm][ql][s];
            mi[s] = TOPI[qm][ql][s];
        }
#pragma unroll
        for (int s = 0; s < 16; s++)
            insert16(mv, mi, TOPV[qm + 2][ql][s], TOPI[qm + 2][ql][s]);
        float* pv = pvals + ((size_t)blockIdx.x * 32 + q) * 16;
        int* pi = pidx + ((size_t)blockIdx.x * 32 + q) * 16;
#pragma unroll
        for (int s = 0; s < 16; s++) {
            pv[s] = mv[s];
            pi[s] = mi[s];
        }
    }
}

// ---------------- Kernel 4: parallel final merge + softmax/statistics ------
__global__ void __launch_bounds__(256) k_merge(const float* __restrict__ pvals,
                                               const int* __restrict__ pidx,
                                               float* __restrict__ out_vals,
                                               float* __restrict__ out_w,
                                               float* __restrict__ out_mean,
                                               float* __restrict__ out_std,
                                               float* __restrict__ out_ent,
                                               int* __restrict__ fidx) {
    __shared__ float MV[8][32][16];
    __shared__ int MI[8][32][16];
    int t = threadIdx.x;
    int q = t & 31, part = t >> 5;

    // stage 1: 8 partitions per query, deterministic chunk order per partition
    float tv[16];
    int ti[16];
#pragma unroll
    for (int s = 0; s < 16; s++) { tv[s] = -__builtin_inff(); ti[s] = 0; }
    for (int c = part; c < N_CHUNKS; c += 8) {
        const float* pv = pvals + ((size_t)c * 32 + q) * 16;
        const int* pi = pidx + ((size_t)c * 32 + q) * 16;
#pragma unroll
        for (int s = 0; s < 16; s++) insert16(tv, ti, pv[s], pi[s]);
    }
#pragma unroll
    for (int s = 0; s < 16; s++) {
        MV[part][q][s] = tv[s];
        MI[part][q][s] = ti[s];
    }
    __syncthreads();

    // stage 2: lane q combines its 8 partitions, sorts, computes stats
    if (t < 32) {
        float mv[16];
        int mi[16];
#pragma unroll
        for (int s = 0; s < 16; s++) {
            mv[s] = MV[0][t][s];
            mi[s] = MI[0][t][s];
        }
        for (int p = 1; p < 8; p++) {
#pragma unroll
            for (int s = 0; s < 16; s++) insert16(mv, mi, MV[p][t][s], MI[p][t][s]);
        }
        // sort descending (unrolled bubble network, constant indices)
#pragma unroll
        for (int a = 0; a < 16; a++)
#pragma unroll
            for (int b = 0; b < 15; b++)
                if (mv[b] < mv[b + 1]) {
                    float fv = mv[b]; mv[b] = mv[b + 1]; mv[b + 1] = fv;
                    int fi = mi[b]; mi[b] = mi[b + 1]; mi[b + 1] = fi;
                }

        float mx = mv[0], se = 0.f, sm = 0.f, sq = 0.f;
        float e[16];
#pragma unroll
        for (int s = 0; s < 16; s++) {
            e[s] = __expf(mv[s] - mx);
            se += e[s];
            sm += mv[s];
            sq += mv[s] * mv[s];
        }
        float inv = 1.0f / se;
        float ent = 0.f;
#pragma unroll
        for (int s = 0; s < 16; s++) {
            float ws = e[s] * inv;
            out_w[t * 16 + s] = ws;
            out_vals[t * 16 + s] = mv[s];
            fidx[t * 16 + s] = mi[s];
            ent -= ws * __logf(fmaxf(ws, 1e-6f));
        }
        float mean = sm * (1.0f / 16.0f);
        float var = sq * (1.0f / 16.0f) - mean * mean;
        out_mean[t] = mean;
        out_std[t] = sqrtf(fmaxf(var, 0.0f));
        out_ent[t] = ent;
    }
}

// ---------------- Kernel 5: gather + renormalize bank rows -----------------
__global__ void k_feats(const float* __restrict__ bank, const int* __restrict__ fidx,
                        float* __restrict__ out_f) {
    __shared__ float red[128];
    int bid = blockIdx.x;  // q*16 + s
    int t = threadIdx.x;   // 128
    int id = fidx[bid];
    const float* row = bank + (size_t)id * 512;
    float s = 0.f;
    for (int k = t; k < 512; k += 128) {
        float x = row[k];
        s += x * x;
    }
    red[t] = s;
    __syncthreads();
    for (int st = 64; st > 0; st >>= 1) {
        if (t < st) red[t] += red[t + st];
        __syncthreads();
    }
    float inv = 1.0f / fmaxf(sqrtf(red[0]), 1e-12f);
    float* o = out_f + (size_t)bid * 512;
    for (int k = t; k < 512; k += 128) o[k] = row[k] * inv;
}

// ---------------- Kernel 6: weighted prototype + l2norm --------------------
__global__ void k_proto(const float* __restrict__ feats, const float* __restrict__ w,
                        float* __restrict__ out_p) {
    __shared__ float red[256];
    int q = blockIdx.x, t = threadIdx.x;
    float a0 = 0.f, a1 = 0.f;
#pragma unroll
    for (int s = 0; s < 16; s++) {
        float ws = w[q * 16 + s];
        const float* f = feats + ((size_t)q * 16 + s) * 512;
        a0 += ws * f[t];
        a1 += ws * f[t + 256];
    }
    red[t] = a0 * a0 + a1 * a1;
    __syncthreads();
    for (int st = 128; st > 0; st >>= 1) {
        if (t < st) red[t] += red[t + st];
        __syncthreads();
    }
    float inv = 1.0f / fmaxf(sqrtf(red[0]), 1e-12f);
    out_p[q * 512 + t] = a0 * inv;
    out_p[q * 512 + t + 256] = a1 * inv;
}

__global__ void k_margin(const float* __restrict__ pm, const float* __restrict__ nm,
                         float* __restrict__ mg) {
    int q = threadIdx.x;
    mg[q] = pm[q] - nm[q];
}

// ---------------------------------------------------------------------------
extern "C" void kernel_launch(void* const* d_in, const int* in_sizes, int n_in,
                              void* d_out, int out_size, void* d_ws, size_t ws_size,
                              hipStream_t stream) {
    (void)in_sizes; (void)n_in; (void)out_size; (void)ws_size;
    const float* qf = (const float*)d_in[0];
    const float* pos = (const float*)d_in[1];
    const float* neg = (const float*)d_in[2];
    const float* W1 = (const float*)d_in[3];
    const float* b1 = (const float*)d_in[4];
    const float* W2 = (const float*)d_in[5];
    const float* b2 = (const float*)d_in[6];
    float* out = (float*)d_out;

    float* qg = out + 0;
    float* pq = out + 16384;
    float* pf = out + 32768;
    float* pw = out + 294912;
    float* ps = out + 295424;
    float* pp = out + 295936;
    float* nf = out + 312320;
    float* nw = out + 574464;
    float* ns = out + 574976;
    float* npr = out + 575488;
    float* pm = out + 591872;
    float* nm = out + 591904;
    float* mg = out + 591936;
    float* psd = out + 591968;
    float* nsd = out + 592000;
    float* pe = out + 592032;
    float* ne = out + 592064;

    float* pvals = (float*)d_ws;                              // [391][32][16] f32
    int* pidx = (int*)(pvals + (size_t)N_CHUNKS * 32 * 16);   // [391][32][16] i32
    int* fxp = pidx + (size_t)N_CHUNKS * 32 * 16;             // [32][16]
    int* fxn = fxp + 32 * 16;                                 // [32][16]

    k_gap<<<32, 256, 0, stream>>>(qf, qg);
    k_mlp<<<1, 256, 0, stream>>>(qg, W1, b1, W2, b2, pq);

    // positive bank
    k_sim<<<N_CHUNKS, 128, 0, stream>>>(pq, pos, pvals, pidx);
    k_merge<<<1, 256, 0, stream>>>(pvals, pidx, ps, pw, pm, psd, pe, fxp);
    k_feats<<<512, 128, 0, stream>>>(pos, fxp, pf);
    k_proto<<<32, 256, 0, stream>>>(pf, pw, pp);

    // negative bank (workspace reused; stream-ordered)
    k_sim<<<N_CHUNKS, 128, 0, stream>>>(pq, neg, pvals, pidx);
    k_merge<<<1, 256, 0, stream>>>(pvals, pidx, ns, nw, nm, nsd, ne, fxn);
    k_feats<<<512, 128, 0, stream>>>(neg, fxn, nf);
    k_proto<<<32, 256, 0, stream>>>(nf, nw, npr);

    k_margin<<<1, 32, 0, stream>>>(pm, nm, mg);
}